// GATConv_45140106281725
// MI455X (gfx1250) — compile-verified
//
#include <hip/hip_runtime.h>

typedef __attribute__((ext_vector_type(2))) float v2f;
typedef __attribute__((ext_vector_type(4))) float v4f;
typedef __attribute__((ext_vector_type(8))) float v8f;

#define IN_C  128
#define HC    128   // HEADS * OUT_C
#define HEADS 4

// ---------------------------------------------------------------------------
// Kernel 1: xp = x @ W   ([N,128] x [128,128]) via V_WMMA_F32_16X16X4_F32
// One wave per 16x16 output tile; K=128 -> 32 WMMA issues.
// A 16x4 / B 4x16 layout: K = kk + 2*(lane>>4) + vgprIdx, M/N = lane&15
// D layout: vgpr v -> row v + 8*(lane>>4), col = lane&15
// ---------------------------------------------------------------------------
__global__ __launch_bounds__(32)
void gat_gemm_wmma(const float* __restrict__ x, const float* __restrict__ w,
                   float* __restrict__ xp, int N) {
  const int lane = threadIdx.x;
  const int half = lane >> 4;
  const int lr   = lane & 15;
  const int tileM = blockIdx.x * 16;
  const int tileN = blockIdx.y * 16;

  int r = tileM + lr;
  if (r >= N) r = N - 1;               // clamp: EXEC must stay all-ones for WMMA
  const float* __restrict__ xrow = x + (size_t)r * IN_C;

  v8f acc = {};
  for (int kk = 0; kk < IN_C; kk += 4) {
    const int kb = kk + 2 * half;
    v2f a, b;
    a[0] = xrow[kb + 0];
    a[1] = xrow[kb + 1];
    b[0] = w[(size_t)(kb + 0) * HC + tileN + lr];
    b[1] = w[(size_t)(kb + 1) * HC + tileN + lr];
    acc = __builtin_amdgcn_wmma_f32_16x16x4_f32(false, a, false, b,
                                                (short)0, acc, false, false);
  }

  float* __restrict__ orow0 = xp + (size_t)(tileM + 8 * half) * HC + tileN + lr;
  if (tileM + 16 <= N) {               // uniform fast path: branch-free stores
#pragma unroll
    for (int v = 0; v < 8; ++v) orow0[(size_t)v * HC] = acc[v];
  } else {
#pragma unroll
    for (int v = 0; v < 8; ++v) {
      const int orow = tileM + v + 8 * half;
      if (orow < N) xp[(size_t)orow * HC + tileN + lr] = acc[v];
    }
  }
}

// ---------------------------------------------------------------------------
// Kernel 2: a_src[n,h] = <xp[n,h,:], att_src[h,:]>, same for a_dst.
// One wave per node (4 nodes/block); lane owns channels 4*lane..4*lane+3,
// head = lane>>3; b128 loads + 8-lane shuffle reduction.
// ---------------------------------------------------------------------------
__global__ __launch_bounds__(128)
void gat_scores(const float* __restrict__ xp, const float* __restrict__ att_src,
                const float* __restrict__ att_dst,
                float* __restrict__ a_src, float* __restrict__ a_dst, int N) {
  const int wave = threadIdx.x >> 5;
  const int lane = threadIdx.x & 31;
  const int n = blockIdx.x * 4 + wave;
  if (n >= N) return;

  const v4f v  = *(const v4f*)(xp + (size_t)n * HC + lane * 4);
  const v4f ws = *(const v4f*)(att_src + lane * 4);
  const v4f wd = *(const v4f*)(att_dst + lane * 4);
  float s = v[0]*ws[0] + v[1]*ws[1] + v[2]*ws[2] + v[3]*ws[3];
  float d = v[0]*wd[0] + v[1]*wd[1] + v[2]*wd[2] + v[3]*wd[3];
#pragma unroll
  for (int off = 4; off > 0; off >>= 1) {   // reduce within the 8-lane head group
    s += __shfl_xor(s, off, 32);
    d += __shfl_xor(d, off, 32);
  }
  if ((lane & 7) == 0) {
    const int h = lane >> 3;
    a_src[n * HEADS + h] = s;
    a_dst[n * HEADS + h] = d;
  }
}

// ---------------------------------------------------------------------------
// Kernel 3: per-destination-node segmented softmax + weighted aggregation.
// One wave per node (4 nodes/block); lane owns channels 4*lane..4*lane+3
// (head = lane>>3) -> b128 gathers of xp[col], b128 stores of out.
// out[n,h,c] = (sum_e p_e * xp[col_e,h,c]) / (sum_e p_e + 1e-8) + bias
// with p_e = exp(leaky_relu(a_src[n,h]+a_dst[col_e,h]) - max(0, seg_max)).
// ---------------------------------------------------------------------------
__device__ __forceinline__ int lb_ll(const long long* __restrict__ a, int n,
                                     long long key) {
  int lo = 0, hi = n;
  while (lo < hi) {
    const int mid = (lo + hi) >> 1;
    if (a[mid] < key) lo = mid + 1; else hi = mid;
  }
  return lo;
}

__global__ __launch_bounds__(128)
void gat_aggregate(const float* __restrict__ xp, const float* __restrict__ a_src,
                   const float* __restrict__ a_dst, const float* __restrict__ bias,
                   const long long* __restrict__ row, const long long* __restrict__ col,
                   float* __restrict__ out, int N, int E) {
  const int wave = threadIdx.x >> 5;
  const int lane = threadIdx.x & 31;
  const int n = blockIdx.x * 4 + wave;
  if (n >= N) return;
  const int h = lane >> 3;               // head owning channels 4*lane..4*lane+3

  const int lo = lb_ll(row, E, (long long)n);
  const int hi = lb_ll(row, E, (long long)n + 1);

  const float as = a_src[n * HEADS + h];

  // pass 1: segment max, seeded with 0 (matches max(0, seg_max) incl. empty seg)
  float m = 0.0f;
  for (int e = lo; e < hi; ++e) {
    const long long c = col[e];
    float al = as + a_dst[(size_t)c * HEADS + h];
    al = (al > 0.0f) ? al : 0.2f * al;
    m = fmaxf(m, al);
  }

  // pass 2: fused exp-sum + weighted gather-accumulate (b128 per lane per edge)
  float psum = 0.0f;
  v4f acc = {};
  for (int e = lo; e < hi; ++e) {
    const long long c = col[e];
    float al = as + a_dst[(size_t)c * HEADS + h];
    al = (al > 0.0f) ? al : 0.2f * al;
    const float p = __expf(al - m);
    psum += p;
    const v4f xv = *(const v4f*)(xp + (size_t)c * HC + lane * 4);
    acc[0] += p * xv[0];
    acc[1] += p * xv[1];
    acc[2] += p * xv[2];
    acc[3] += p * xv[3];
  }

  const float inv = 1.0f / (psum + 1e-8f);
  const v4f b = *(const v4f*)(bias + lane * 4);
  v4f o;
  o[0] = acc[0] * inv + b[0];
  o[1] = acc[1] * inv + b[1];
  o[2] = acc[2] * inv + b[2];
  o[3] = acc[3] * inv + b[3];
  *(v4f*)(out + (size_t)n * HC + lane * 4) = o;
}

// ---------------------------------------------------------------------------
extern "C" void kernel_launch(void* const* d_in, const int* in_sizes, int n_in,
                              void* d_out, int out_size, void* d_ws, size_t ws_size,
                              hipStream_t stream) {
  const float*     x       = (const float*)d_in[0];
  const float*     weight  = (const float*)d_in[1];
  const float*     att_src = (const float*)d_in[2];
  const float*     att_dst = (const float*)d_in[3];
  const float*     bias    = (const float*)d_in[4];
  const long long* row     = (const long long*)d_in[5];
  const long long* col     = (const long long*)d_in[6];
  float*           out     = (float*)d_out;

  const int N = in_sizes[0] / IN_C;
  const int E = in_sizes[5];

  float* xp    = (float*)d_ws;                    // [N,128]
  float* a_src = xp + (size_t)N * HC;             // [N,4]
  float* a_dst = a_src + (size_t)N * HEADS;       // [N,4]

  dim3 g1((N + 15) / 16, HC / 16);
  gat_gemm_wmma<<<g1, 32, 0, stream>>>(x, weight, xp, N);
  const int nb = (N + 3) / 4;                     // 4 nodes (waves) per block
  gat_scores<<<nb, 128, 0, stream>>>(xp, att_src, att_dst, a_src, a_dst, N);
  gat_aggregate<<<nb, 128, 0, stream>>>(xp, a_src, a_dst, bias, row, col, out, N, E);
}